// GCN_33363305955328
// MI455X (gfx1250) — compile-verified
//
#include <hip/hip_runtime.h>

typedef __attribute__((ext_vector_type(2))) float v2f;
typedef __attribute__((ext_vector_type(8))) float v8f;

#define N_NODES 100000
#define N_EDGES 3200000
#define F_IN    256
#define CH      16
#define FCN     64
#define NTILES  (N_NODES / 16)   // 6250 tiles of 16 nodes (N divisible by 16)

// ---------------------------------------------------------------------------
// Kernel 1: h = x @ w_gcn   ([100000,256] x [256,16])
// One wave per 16-node tile; full K=256 as 64 chained V_WMMA_F32_16X16X4_F32.
// A frag (16x4 f32): lanes 0-15 hold K={kk,kk+1}, lanes 16-31 hold K={kk+2,kk+3}.
// B frag (4x16 f32): mirrored K split, N = lane%16.  B sourced from LDS copy of w_gcn.
// D (16x16 f32, 8 VGPRs): VGPR j -> row j (lanes 0-15) / row j+8 (lanes 16-31).
// ---------------------------------------------------------------------------
__global__ __launch_bounds__(256) void gcn_feat_wmma(
    const float* __restrict__ x, const float* __restrict__ wg,
    float* __restrict__ h)
{
    __shared__ float lds_w[F_IN * CH];            // 16 KB
    const int tid = threadIdx.x;
    for (int i = tid; i < F_IN * CH; i += 256) lds_w[i] = wg[i];
    __syncthreads();

    const int wave = tid >> 5;
    const int lane = tid & 31;
    const int tile = blockIdx.x * 8 + wave;
    if (tile >= NTILES) return;                   // wave-uniform; EXEC stays all-ones

    const int m  = lane & 15;
    const int hi = lane >> 4;                     // 0: K pair {0,1}; 1: K pair {2,3}
    const int row0 = tile * 16;
    const float* arow = x + (size_t)(row0 + m) * F_IN + 2 * hi;

    v8f acc = {0.f, 0.f, 0.f, 0.f, 0.f, 0.f, 0.f, 0.f};
    #pragma unroll 8
    for (int k = 0; k < 64; ++k) {
        const int kk = 4 * k;
        v2f a; a.x = arow[kk];     a.y = arow[kk + 1];         // global_load_b64, row-sequential
        v2f b; b.x = lds_w[(kk + 2 * hi)     * CH + m];
               b.y = lds_w[(kk + 2 * hi + 1) * CH + m];
        acc = __builtin_amdgcn_wmma_f32_16x16x4_f32(
                  false, a, false, b, (short)0, acc, false, false);
    }

    float* hrow = h + (size_t)row0 * CH + m;
    #pragma unroll
    for (int j = 0; j < 8; ++j)
        hrow[(size_t)(j + 8 * hi) * CH] = acc[j];
}

// ---------------------------------------------------------------------------
// Kernel 2: msgs = h[src]*edge_w ; agg[dst] += msgs    (scatter-add, L2 atomics)
// Thread = (edge, channel): lane%16 = channel -> 64B coalesced gather + atomics.
// ---------------------------------------------------------------------------
__global__ __launch_bounds__(256) void gcn_edge_scatter(
    const float* __restrict__ h, const int* __restrict__ src,
    const int* __restrict__ dst, const float* __restrict__ ew,
    float* __restrict__ agg)
{
    const long long idx = (long long)blockIdx.x * 256 + threadIdx.x;
    if (idx >= (long long)N_EDGES * CH) return;
    const int e = (int)(idx >> 4);
    const int c = (int)(idx & 15);
    const int s = src[e];
    const int d = dst[e];
    const float v = h[(size_t)s * CH + c] * ew[e];
    atomicAdd(&agg[(size_t)d * CH + c], v);       // global_atomic_add_f32, L2-resident
}

// ---------------------------------------------------------------------------
// Kernel 3: o1 = relu(relu(agg) @ w0 + b0) ; out = o1 @ w1 + b1
// WMMA: 16 nodes x 64 FCN = 4 column tiles, K=16 = 4 x4-steps (relu fused into A).
// o1 staged in LDS (padded), then 2 half-row dots/lane + shfl_xor(16) reduction.
// ---------------------------------------------------------------------------
__global__ __launch_bounds__(256) void gcn_head_wmma(
    const float* __restrict__ agg, const float* __restrict__ w0,
    const float* __restrict__ b0,  const float* __restrict__ w1,
    const float* __restrict__ b1,  float* __restrict__ out)
{
    __shared__ float lds_w0[CH * 80];             // row stride 80: halves hit disjoint banks
    __shared__ float lds_b0[FCN];
    __shared__ float lds_w1[FCN];
    __shared__ float lds_o1[8][16 * 68];          // per-wave o1, node stride 68

    const int tid = threadIdx.x;
    for (int i = tid; i < CH * FCN; i += 256)
        lds_w0[(i / FCN) * 80 + (i % FCN)] = w0[i];
    if (tid < FCN) { lds_b0[tid] = b0[tid]; lds_w1[tid] = w1[tid]; }
    __syncthreads();

    const int wave = tid >> 5;
    const int lane = tid & 31;
    const int tile = blockIdx.x * 8 + wave;
    if (tile >= NTILES) return;

    const int m  = lane & 15;
    const int hi = lane >> 4;
    const int row0 = tile * 16;
    const float* arow = agg + (size_t)(row0 + m) * CH + 2 * hi;

    v8f acc[4];
    #pragma unroll
    for (int t = 0; t < 4; ++t) acc[t] = (v8f){0.f,0.f,0.f,0.f,0.f,0.f,0.f,0.f};

    #pragma unroll
    for (int k = 0; k < 4; ++k) {
        const int kk = 4 * k;
        v2f a; a.x = fmaxf(arow[kk], 0.f);        // relu(agg) fused into A fragment
               a.y = fmaxf(arow[kk + 1], 0.f);
        const float* br0 = &lds_w0[(kk + 2 * hi)     * 80 + m];
        const float* br1 = &lds_w0[(kk + 2 * hi + 1) * 80 + m];
        #pragma unroll
        for (int t = 0; t < 4; ++t) {
            v2f b; b.x = br0[16 * t]; b.y = br1[16 * t];
            acc[t] = __builtin_amdgcn_wmma_f32_16x16x4_f32(
                         false, a, false, b, (short)0, acc[t], false, false);
        }
    }

    // bias + relu, stage o1 rows into LDS (wave-private region)
    float* o1 = &lds_o1[wave][0];
    #pragma unroll
    for (int t = 0; t < 4; ++t) {
        const float bias = lds_b0[16 * t + m];
        #pragma unroll
        for (int j = 0; j < 8; ++j)
            o1[(j + 8 * hi) * 68 + 16 * t + m] = fmaxf(acc[t][j] + bias, 0.f);
    }

    // out[node] = o1[node,:] . w1 + b1 : each lane sums 32 of 64, combine via xor-16
    const float* row = &lds_o1[wave][m * 68 + 32 * hi];
    const float* wv  = &lds_w1[32 * hi];
    float s = 0.f;
    #pragma unroll
    for (int f = 0; f < 32; ++f) s += row[f] * wv[f];
    s += __shfl_xor(s, 16, 32);
    if (hi == 0) out[row0 + m] = s + b1[0];
}

// ---------------------------------------------------------------------------
extern "C" void kernel_launch(void* const* d_in, const int* in_sizes, int n_in,
                              void* d_out, int out_size, void* d_ws, size_t ws_size,
                              hipStream_t stream)
{
    (void)in_sizes; (void)n_in; (void)out_size; (void)ws_size;
    const float* x  = (const float*)d_in[0];
    const int*   ei = (const int*)  d_in[1];   // [2, E] row-major: src then dst
    const float* ew = (const float*)d_in[2];
    const float* wg = (const float*)d_in[3];
    const float* w0 = (const float*)d_in[4];
    const float* b0 = (const float*)d_in[5];
    const float* w1 = (const float*)d_in[6];
    const float* b1 = (const float*)d_in[7];
    float* out = (float*)d_out;

    float* h   = (float*)d_ws;                         // N*CH f32 (6.4 MB)
    float* agg = h + (size_t)N_NODES * CH;             // N*CH f32 (6.4 MB)

    // agg must be zeroed every call (workspace is not re-poisoned between replays)
    hipMemsetAsync(agg, 0, (size_t)N_NODES * CH * sizeof(float), stream);

    const int tile_blocks = (NTILES + 7) / 8;          // 8 waves (tiles) per 256-thread block
    gcn_feat_wmma<<<tile_blocks, 256, 0, stream>>>(x, wg, h);

    const long long ework = (long long)N_EDGES * CH;
    const int edge_blocks = (int)((ework + 255) / 256);
    gcn_edge_scatter<<<edge_blocks, 256, 0, stream>>>(h, ei, ei + N_EDGES, ew, agg);

    gcn_head_wmma<<<tile_blocks, 256, 0, stream>>>(agg, w0, b0, w1, b1, out);
}